// DeformConv1d_55490977464554
// MI455X (gfx1250) — compile-verified
//
#include <hip/hip_runtime.h>
#include <math.h>

// ---- static problem config (matches reference) ----
#define BB    8
#define CIN   64
#define COUT  64
#define LL    32768
#define KK    5
#define PADC  2
#define GG    4
#define DGG   2
#define GDG   8      // GG*DGG
#define CD    8      // CIN/GDG
#define DWK   7
#define OC    40     // GDG*KK  (offset/mask rows)
#define LOUTC 32768
#define EPSF  1e-5f

#define TILE  64     // columns per branch-kernel workgroup
#define XS    72     // LDS stride for x tile (needs 70)
#define HS    72     // LDS stride for h  (bank-disjoint half-wave reads)
#define OS    68     // LDS stride for GEMM out
#define WS3   84     // LDS stride for deform weights (needs 80)
#define NT3   128    // columns per deform-kernel workgroup (8 waves x 16)
#define SS3   144    // LDS stride for sampling params (disjoint bank spans)

typedef __attribute__((ext_vector_type(2))) float v2f;
typedef __attribute__((ext_vector_type(8))) float v8f;

__device__ __forceinline__ v8f wmma_f32_16x16x4(v2f a, v2f b, v8f c) {
  // D = A(16x4,f32) * B(4x16,f32) + C(16x16,f32)
  return __builtin_amdgcn_wmma_f32_16x16x4_f32(false, a, false, b, (short)0, c,
                                               false, false);
}

// ---------------------------------------------------------------------------
// Kernel 1: per-(b,c) mean / rsqrt(var) of the depthwise-conv output for both
// branches (GroupNorm with groups == channels reduces over L only).
// ---------------------------------------------------------------------------
__global__ __launch_bounds__(256)
void dc_stats_kernel(const float* __restrict__ x,
                     const float* __restrict__ dwo_w, const float* __restrict__ dwo_b,
                     const float* __restrict__ dwm_w, const float* __restrict__ dwm_b,
                     float* __restrict__ stats) {
  const int bc  = blockIdx.x;        // b*CIN + c
  const int c   = bc & (CIN - 1);
  const int tid = threadIdx.x;
  const float* xr = x + (size_t)bc * LL;

  float wo[DWK], wm[DWK];
#pragma unroll
  for (int j = 0; j < DWK; ++j) { wo[j] = dwo_w[c*DWK + j]; wm[j] = dwm_w[c*DWK + j]; }
  const float bo = dwo_b[c], bm = dwm_b[c];

  float so = 0.f, qo = 0.f, sm = 0.f, qm = 0.f;
  for (int l = tid; l < LL; l += 256) {
    float yo = bo, ym = bm;
#pragma unroll
    for (int j = 0; j < DWK; ++j) {
      int xl = l - (DWK - 1) / 2 + j;
      float v = (xl >= 0 && xl < LL) ? xr[xl] : 0.f;
      yo += wo[j] * v; ym += wm[j] * v;
    }
    so += yo; qo += yo * yo; sm += ym; qm += ym * ym;
  }

  __shared__ float red[4][256];
  red[0][tid] = so; red[1][tid] = qo; red[2][tid] = sm; red[3][tid] = qm;
  __syncthreads();
  for (int s = 128; s > 0; s >>= 1) {
    if (tid < s) {
      red[0][tid] += red[0][tid + s]; red[1][tid] += red[1][tid + s];
      red[2][tid] += red[2][tid + s]; red[3][tid] += red[3][tid + s];
    }
    __syncthreads();
  }
  if (tid == 0) {
    const float inv = 1.f / (float)LL;
    float mo = red[0][0] * inv, vo = red[1][0] * inv - mo * mo;
    float mm = red[2][0] * inv, vm = red[3][0] * inv - mm * mm;
    stats[bc*4 + 0] = mo; stats[bc*4 + 1] = rsqrtf(vo + EPSF);
    stats[bc*4 + 2] = mm; stats[bc*4 + 3] = rsqrtf(vm + EPSF);
  }
}

// ---------------------------------------------------------------------------
// Kernel 2: fused branch pass. Per (b, 64-col tile): stage x (+halo) in LDS,
// compute dwconv+GroupNorm h for all 64 channels, then 48x64 x 64x64 GEMM via
// V_WMMA_F32_16X16X4_F32 (rows 40..47 zero-padded). Branch 0 -> offset*4,
// branch 1 -> softmax over the 5 taps of each deformable group.
// ---------------------------------------------------------------------------
__global__ __launch_bounds__(256)
void dc_branch_kernel(const float* __restrict__ x,
                      const float* __restrict__ dwo_w, const float* __restrict__ dwo_b,
                      const float* __restrict__ gno_w, const float* __restrict__ gno_b,
                      const float* __restrict__ pwo_w, const float* __restrict__ pwo_b,
                      const float* __restrict__ dwm_w, const float* __restrict__ dwm_b,
                      const float* __restrict__ gnm_w, const float* __restrict__ gnm_b,
                      const float* __restrict__ pwm_w, const float* __restrict__ pwm_b,
                      const float* __restrict__ stats,
                      float* __restrict__ offsb, float* __restrict__ attnb) {
  __shared__ float sx[CIN * XS];     // x tile with +/-3 halo
  __shared__ float hbuf[CIN * HS];   // normalized dwconv output (GEMM B operand)
  __shared__ float wbuf[48 * 64];    // zero-padded pointwise weights (A operand)
  __shared__ float obuf[48 * OS];    // GEMM result staging

  const int b    = blockIdx.y;
  const int l0   = blockIdx.x * TILE;
  const int tid  = threadIdx.x;
  const int lane = tid & 31;
  // scalarize the wave index so the WMMA tile loop is SGPR-controlled and
  // EXEC stays all-ones around the matrix ops (ISA requirement).
  const int wv   = __builtin_amdgcn_readfirstlane(tid >> 5);
  const int hl   = lane >> 4;        // half-wave select (K pair)
  const int q    = lane & 15;        // row (A) / col (B,C) within fragment

  for (int idx = tid; idx < CIN * (TILE + DWK - 1); idx += 256) {
    int c = idx / (TILE + DWK - 1), j = idx % (TILE + DWK - 1);
    int gl = l0 - (DWK - 1) / 2 + j;
    sx[c*XS + j] = (gl >= 0 && gl < LL) ? x[(size_t)(b*CIN + c)*LL + gl] : 0.f;
  }
  __syncthreads();

  for (int br = 0; br < 2; ++br) {
    const float* dw_w = br ? dwm_w : dwo_w;
    const float* dw_b = br ? dwm_b : dwo_b;
    const float* gn_w = br ? gnm_w : gno_w;
    const float* gn_b = br ? gnm_b : gno_b;
    const float* pw_w = br ? pwm_w : pwo_w;
    const float* pw_b = br ? pwm_b : pwo_b;
    const int    st   = br ? 2 : 0;

    // depthwise conv + per-channel GroupNorm -> hbuf[channel][col]
    for (int idx = tid; idx < CIN * TILE; idx += 256) {
      int c = idx >> 6, l = idx & 63;
      float acc = dw_b[c];
#pragma unroll
      for (int j = 0; j < DWK; ++j) acc += dw_w[c*DWK + j] * sx[c*XS + l + j];
      float mu = stats[(b*CIN + c)*4 + st];
      float rs = stats[(b*CIN + c)*4 + st + 1];
      hbuf[c*HS + l] = (acc - mu) * rs * gn_w[c] + gn_b[c];
    }
    // stage pointwise weights, zero-pad rows 40..47
    for (int idx = tid; idx < 48 * 64; idx += 256) {
      int r = idx >> 6, cc = idx & 63;
      wbuf[idx] = (r < OC) ? pw_w[r*CIN + cc] : 0.f;
    }
    __syncthreads();

    // 12 output tiles (3 M-tiles x 4 N-tiles) over 8 waves; t is scalar.
    for (int t = wv; t < 12; t += 8) {
      const int mt = (t % 3) * 16;
      const int nt = (t / 3) * 16;
      v8f acc = {};
#pragma unroll 4
      for (int kk = 0; kk < 64; kk += 4) {
        const int kb = kk + 2 * hl;
        v2f a, bv;
        a.x  = wbuf[(mt + q)*64 + kb];
        a.y  = wbuf[(mt + q)*64 + kb + 1];
        bv.x = hbuf[kb*HS + nt + q];
        bv.y = hbuf[(kb + 1)*HS + nt + q];
        acc = wmma_f32_16x16x4(a, bv, acc);
      }
#pragma unroll
      for (int r = 0; r < 8; ++r)
        obuf[(mt + r + 8*hl)*OS + nt + q] = acc[r];
    }
    __syncthreads();

    if (br == 0) {
      for (int idx = tid; idx < OC * TILE; idx += 256) {
        int o = idx >> 6, l = idx & 63;
        offsb[(size_t)(b*OC + o)*LOUTC + l0 + l] = (obuf[o*OS + l] + pw_b[o]) * 4.0f;
      }
    } else {
      for (int idx = tid; idx < GDG * TILE; idx += 256) {
        int gd = idx >> 6, l = idx & 63;
        float v[KK]; float mx = -3.0e38f;
#pragma unroll
        for (int k = 0; k < KK; ++k) {
          v[k] = obuf[(gd*KK + k)*OS + l] + pw_b[gd*KK + k];
          mx = fmaxf(mx, v[k]);
        }
        float sum = 0.f;
#pragma unroll
        for (int k = 0; k < KK; ++k) { v[k] = __expf(v[k] - mx); sum += v[k]; }
        float inv = 1.f / sum;
#pragma unroll
        for (int k = 0; k < KK; ++k)
          attnb[(size_t)(b*OC + gd*KK + k)*LOUTC + l0 + l] = v[k] * inv;
      }
    }
    __syncthreads();
  }
}

// ---------------------------------------------------------------------------
// Kernel 3: deformable conv. Per (b,g), out[16,L] = W[16,80] x S[80,L] + bias.
// Stage 1 (cooperative): per (d*5+k, col) precompute the clamped gather index
// and fused interp*attn*validity coefficients a0/a1 -- this removes the 8x
// per-channel redundancy of the sampling math (the kernel is VALU-bound).
// Stage 2: each wave owns a 16-col tile; 20 fp32 WMMA steps, B-fragment built
// from 2 global gathers + 2 FMAs per element.
// ---------------------------------------------------------------------------
__global__ __launch_bounds__(256)
void dc_deform_kernel(const float* __restrict__ x,
                      const float* __restrict__ weight, const float* __restrict__ bias,
                      const float* __restrict__ offsb, const float* __restrict__ attnb,
                      float* __restrict__ out) {
  __shared__ float wl[16 * WS3];          // [o][dck], dck = d*40 + c*5 + k
  __shared__ int   si0[DGG*KK * SS3];     // clamped gather index per (dk, col)
  __shared__ float sa0[DGG*KK * SS3];     // (1-w1)*attn*valid0
  __shared__ float sa1[DGG*KK * SS3];     // w1*attn*valid1

  const int b    = blockIdx.y >> 2;
  const int g    = blockIdx.y & 3;
  const int tid  = threadIdx.x;
  const int lane = tid & 31;
  const int wv   = __builtin_amdgcn_readfirstlane(tid >> 5);
  const int hl   = lane >> 4;
  const int q    = lane & 15;
  const int c0   = blockIdx.x * NT3;      // first column of this block

  // stage grouped weight as [o(16)][dck(80)]
  for (int idx = tid; idx < 16 * 80; idx += 256) {
    int o = idx / 80, r = idx % 80;
    int d = r / 40, rem = r % 40, c = rem / KK, k = rem % KK;
    wl[o*WS3 + r] = weight[((size_t)(g*16 + o)*16 + (d*CD + c))*KK + k];
  }

  const float* xb   = x     + (size_t)(b*CIN + g*16) * LL;        // + (d*8+c)*LL
  const float* offg = offsb + (size_t)(b*OC + g*DGG*KK) * LOUTC;  // row = d*5+k
  const float* attg = attnb + (size_t)(b*OC + g*DGG*KK) * LOUTC;

  // precompute sampling params: 10 rows x 128 cols, once per (d,k,col)
  for (int idx = tid; idx < DGG*KK * NT3; idx += 256) {
    int dk = idx >> 7, j = idx & (NT3 - 1);
    int k  = dk % KK;
    int col = c0 + j;
    float off = offg[(size_t)dk*LOUTC + col];
    float at  = attg[(size_t)dk*LOUTC + col];
    float p   = (float)(col - PADC + k) + off;
    float p0  = floorf(p);
    float w1  = p - p0;
    int   i0  = (int)p0;
    float a0  = (i0 >= 0  && i0 <  LL)     ? (1.f - w1) * at : 0.f;
    float a1  = (i0 >= -1 && i0 <  LL - 1) ? w1 * at         : 0.f;
    int   i0c = i0 < 0 ? 0 : (i0 > LL - 2 ? LL - 2 : i0);
    si0[dk*SS3 + j] = i0c;
    sa0[dk*SS3 + j] = a0;
    sa1[dk*SS3 + j] = a1;
  }
  __syncthreads();

  const int jj  = wv * 16 + q;            // column within block (per wave tile)
  const int col = c0 + jj;

  v8f acc = {};
#pragma unroll 5
  for (int kk = 0; kk < 80; kk += 4) {
    const int r0 = kk + 2 * hl;
    v2f a, bv;
    a.x = wl[q*WS3 + r0];
    a.y = wl[q*WS3 + r0 + 1];
#pragma unroll
    for (int u = 0; u < 2; ++u) {
      int r = r0 + u;
      int d = r / 40, rem = r % 40, c = rem / KK, k = rem % KK;
      int dk = d*KK + k;
      int   i0 = si0[dk*SS3 + jj];
      float a0 = sa0[dk*SS3 + jj];
      float a1 = sa1[dk*SS3 + jj];
      const float* xr = xb + (size_t)(d*CD + c) * LL;
      float s = xr[i0] * a0 + xr[i0 + 1] * a1;
      if (u == 0) bv.x = s; else bv.y = s;
    }
    acc = wmma_f32_16x16x4(a, bv, acc);
  }
#pragma unroll
  for (int r = 0; r < 8; ++r) {
    int m = r + 8 * hl;
    out[(size_t)(b*COUT + g*16 + m)*LOUTC + col] = acc[r] + bias[g*16 + m];
  }
}

// ---------------------------------------------------------------------------
extern "C" void kernel_launch(void* const* d_in, const int* in_sizes, int n_in,
                              void* d_out, int out_size, void* d_ws, size_t ws_size,
                              hipStream_t stream) {
  const float* x      = (const float*)d_in[0];
  const float* dwo_w  = (const float*)d_in[1];
  const float* dwo_b  = (const float*)d_in[2];
  const float* gno_w  = (const float*)d_in[3];
  const float* gno_b  = (const float*)d_in[4];
  const float* pwo_w  = (const float*)d_in[5];
  const float* pwo_b  = (const float*)d_in[6];
  const float* dwm_w  = (const float*)d_in[7];
  const float* dwm_b  = (const float*)d_in[8];
  const float* gnm_w  = (const float*)d_in[9];
  const float* gnm_b  = (const float*)d_in[10];
  const float* pwm_w  = (const float*)d_in[11];
  const float* pwm_b  = (const float*)d_in[12];
  const float* weight = (const float*)d_in[13];
  const float* bias   = (const float*)d_in[14];
  float* out = (float*)d_out;

  float* stats = (float*)d_ws;                         //  8KB
  float* offsb = stats + BB*CIN*4;                     // ~42MB
  float* attnb = offsb + (size_t)BB*OC*LOUTC;          // ~42MB

  dc_stats_kernel<<<dim3(BB*CIN), dim3(256), 0, stream>>>(
      x, dwo_w, dwo_b, dwm_w, dwm_b, stats);
  dc_branch_kernel<<<dim3(LL/TILE, BB), dim3(256), 0, stream>>>(
      x, dwo_w, dwo_b, gno_w, gno_b, pwo_w, pwo_b,
      dwm_w, dwm_b, gnm_w, gnm_b, pwm_w, pwm_b, stats, offsb, attnb);
  dc_deform_kernel<<<dim3(LOUTC/NT3, BB*GG), dim3(256), 0, stream>>>(
      x, weight, bias, offsb, attnb, out);

  (void)in_sizes; (void)n_in; (void)out_size; (void)ws_size;
}